// OptimalControl_3917010174730
// MI455X (gfx1250) — compile-verified
//
#include <hip/hip_runtime.h>

typedef __attribute__((ext_vector_type(2))) float v2f;
typedef __attribute__((ext_vector_type(8))) float v8f;

#define H 128
#define S 1024
#define NA 512
#define ROWSTRIDE 516                 // padded floats per s-row in LDS (conflict-free ds_load_b64)
#define SLICE_FLOATS (16 * ROWSTRIDE) // one k-slice buffer (16 s-rows x 512 a, padded)
#define LDS_BYTES (2 * SLICE_FLOATS * 4)

// Zero-fill d_out before split-K atomic accumulation (harness poisons it).
__global__ void oc_zero_kernel(float* __restrict__ out, int n) {
  int i = blockIdx.x * blockDim.x + threadIdx.x;
  if (i < n) out[i] = 0.0f;
}

__device__ __forceinline__ v8f wmma_f32(v2f a, v2f b, v8f c) {
  return __builtin_amdgcn_wmma_f32_16x16x4_f32(false, a, false, b, (short)0, c,
                                               false, false);
}

// Compute one k-slice: 128 WMMAs fed by LDS (B) and L1-resident us (A-operand).
template <bool MASKED>
__device__ __forceinline__ void compute_slice(const float* __restrict__ usrow,
                                              const float* __restrict__ lrow,
                                              bool valid, v8f& acc0, v8f& acc1) {
#pragma unroll 4
  for (int a = 0; a < NA; a += 8) {
    v2f av0 = *(const v2f*)(usrow + a);
    v2f bv0 = *(const v2f*)(lrow + a);
    v2f av1 = *(const v2f*)(usrow + a + 4);
    v2f bv1 = *(const v2f*)(lrow + a + 4);
    if (MASKED && !valid) {
      av0.x = 0.0f; av0.y = 0.0f;
      av1.x = 0.0f; av1.y = 0.0f;
    }
    acc0 = wmma_f32(av0, bv0, acc0);   // dual accumulators: independent chains
    acc1 = wmma_f32(av1, bv1, acc1);
  }
}

// One workgroup (8 waves = 8 t-tiles) per (s-strip, k-chunk).
//   blockIdx.x = strip : s0 = strip*16   (64 strips)
//   blockIdx.y = c     : k in [16c,16c+16) (8 chunks)
// Wave w computes t-tile t0 = 16w iff w >= c; ALL waves cooperatively stage
// each 32KB A k-slice into LDS exactly once via async global->LDS copies.
__global__ __launch_bounds__(256)
void oc_conv_wmma_kernel(const float* __restrict__ us,
                         const float* __restrict__ A,
                         float* __restrict__ out) {
  extern __shared__ float lds[];

  const int s0   = blockIdx.x * 16;
  const int c    = blockIdx.y;
  const int k0   = c * 16;
  const int tid  = threadIdx.x;
  const int w    = tid >> 5;          // wave id == t-tile index
  const int lane = tid & 31;
  const int half = lane >> 4;         // K pair select / M-row half
  const int m    = lane & 15;         // M row (t) for A-op; N col (s) for B/C/D
  const int t0   = w * 16;
  const bool active = (w >= c);

  // ---- async staging: each thread moves 8 x 16B per slice (32KB / 256 thr) ----
  const float* gslice0 = A + ((size_t)k0 * S + s0) * NA;
  auto issue_slice = [&](int kk, int buf) {
    const float* gbase = gslice0 + (size_t)kk * (S * NA);
    unsigned lbase = (unsigned)(uintptr_t)lds + (unsigned)(buf * SLICE_FLOATS * 4);
#pragma unroll
    for (int i = 0; i < 8; ++i) {
      int f   = (i * 256 + tid) * 4;  // flat float index in 16x512 slice
      int row = f >> 9;               // s row (0..15)
      int col = f & 511;              // a col (multiple of 4)
      const float* gaddr = gbase + (size_t)row * NA + col;
      unsigned laddr = lbase + (unsigned)(row * ROWSTRIDE + col) * 4;
      asm volatile("global_load_async_to_lds_b128 %0, %1, off"
                   :: "v"(laddr), "v"(gaddr) : "memory");
    }
  };

  issue_slice(0, 0);                  // prologue

  v8f acc0 = {}, acc1 = {};

  for (int kk = 0; kk < 16; ++kk) {
    const int buf = kk & 1;
    asm volatile("s_wait_asynccnt 0x0" ::: "memory");  // own copies landed in LDS
    __syncthreads();                                   // slice kk visible to all
    if (kk + 1 < 16) issue_slice(kk + 1, buf ^ 1);     // prefetch next slice

    if (active) {
      const int k = k0 + kk;
      const int j = t0 + m - k;                        // Toeplitz us row
      const bool valid = (j >= 0);                     // false only when w == c
      const float* usrow = us + (size_t)(valid ? j : 0) * NA + 2 * half;
      const float* lrow  = lds + buf * SLICE_FLOATS + m * ROWSTRIDE + 2 * half;
      if (w > c) compute_slice<false>(usrow, lrow, true,  acc0, acc1);
      else       compute_slice<true >(usrow, lrow, valid, acc0, acc1);
    }
  }

  // Lane l, reg r -> t = t0 + r + 8*half, s = s0 + m. Split-K partials: f32 atomics.
  if (active) {
    float* obase = out + (size_t)(t0 + 8 * half) * S + (s0 + m);
#pragma unroll
    for (int r = 0; r < 8; ++r) {
      __hip_atomic_fetch_add(obase + (size_t)r * S, acc0[r] + acc1[r],
                             __ATOMIC_RELAXED, __HIP_MEMORY_SCOPE_AGENT);
    }
  }
}

extern "C" void kernel_launch(void* const* d_in, const int* in_sizes, int n_in,
                              void* d_out, int out_size, void* d_ws, size_t ws_size,
                              hipStream_t stream) {
  const float* us = (const float*)d_in[0];   // [H, NA]
  const float* A  = (const float*)d_in[1];   // [H, S, NA]
  float* out = (float*)d_out;                // [H, S]

  oc_zero_kernel<<<(H * S + 255) / 256, 256, 0, stream>>>(out, H * S);

  dim3 grid(S / 16, H / 16);                 // (64 strips, 8 k-chunks)
  oc_conv_wmma_kernel<<<grid, 256, LDS_BYTES, stream>>>(us, A, out);
}